// NetW1_66030827209287
// MI455X (gfx1250) — compile-verified
//
#include <hip/hip_runtime.h>
#include <math.h>

typedef __attribute__((ext_vector_type(2))) float v2f;
typedef __attribute__((ext_vector_type(8))) float v8f;

#define NROWS 4194304
#define NWG   1024
#define TPB   256
#define ITERS (NROWS / (NWG * TPB))   // 16

// -----------------------------------------------------------------------------
// Stage kernel.
//   MODE 0: moments of src only (used for x)
//   MODE 1: h = relu(A1*a + c1); moments of h; no store (h recomputed later)
//   MODE 2: h = relu(A1*a + c1); z = relu(A2*h + c2 + SW*a + SB);
//           store z (SoA, possibly in-place over src); moments of z
// Moments S = sum_rows [v;1][v;1]^T (6x6) accumulated with
// V_WMMA_F32_16X16X4_F32 (M=N=feat padded to 16, K=4 batch rows; A==B frag
// since the product is symmetric). Per-WG partial sums written deterministically.
// -----------------------------------------------------------------------------
template <int MODE, bool ROWMAJOR>
__global__ __launch_bounds__(TPB) void stage_kernel(
    const float* __restrict__ src, float* __restrict__ dst,
    const float* __restrict__ p1, const float* __restrict__ p2,
    const float* __restrict__ skW, const float* __restrict__ skB,
    float* __restrict__ partials)
{
  __shared__ float zb[8][32][17];   // row stride 17 -> bank-conflict free
  __shared__ float wsum[8][36];

  const int tid  = threadIdx.x;
  const int wv   = tid >> 5;
  const int lane = tid & 31;
  const int hi   = lane >> 4;       // which K-pair (0 -> k=0,1 ; 1 -> k=2,3)
  const int lm   = lane & 15;       // feature index for A/B fragment

  // zero-pad features 5..15 (slot 5 = constant 1 for mean/count row)
#pragma unroll
  for (int f = 5; f < 16; ++f) zb[wv][lane][f] = (f == 5) ? 1.0f : 0.0f;

  float A1[25], c1[5], A2[25], c2[5], SW[25], SB[5];
  if (MODE >= 1) {
#pragma unroll
    for (int i = 0; i < 25; ++i) A1[i] = p1[i];
#pragma unroll
    for (int i = 0; i < 5; ++i) c1[i] = p1[25 + i];
  }
  if (MODE == 2) {
#pragma unroll
    for (int i = 0; i < 25; ++i) { A2[i] = p2[i]; SW[i] = skW[i]; }
#pragma unroll
    for (int i = 0; i < 5; ++i)  { c2[i] = p2[25 + i]; SB[i] = skB[i]; }
  }
  asm volatile("s_wait_dscnt 0x0" ::: "memory");

  v8f acc = {};

  for (int it = 0; it < ITERS; ++it) {
    const size_t r = (size_t)it * (NWG * TPB) + (size_t)blockIdx.x * TPB + tid;

    float a[5];
    if (ROWMAJOR) {
#pragma unroll
      for (int f = 0; f < 5; ++f) a[f] = src[r * 5 + f];
    } else {
#pragma unroll
      for (int f = 0; f < 5; ++f) a[f] = src[(size_t)f * NROWS + r];
    }

    float vput[5];
    if (MODE == 0) {
#pragma unroll
      for (int f = 0; f < 5; ++f) vput[f] = a[f];
    } else {
      float h[5];
#pragma unroll
      for (int i = 0; i < 5; ++i) {
        float p = c1[i];
#pragma unroll
        for (int j = 0; j < 5; ++j) p = fmaf(A1[i * 5 + j], a[j], p);
        h[i] = fmaxf(p, 0.0f);
      }
      if (MODE == 1) {
#pragma unroll
        for (int f = 0; f < 5; ++f) vput[f] = h[f];
      } else {
#pragma unroll
        for (int i = 0; i < 5; ++i) {
          float p = c2[i] + SB[i];
#pragma unroll
          for (int j = 0; j < 5; ++j)
            p = fmaf(A2[i * 5 + j], h[j], fmaf(SW[i * 5 + j], a[j], p));
          vput[i] = fmaxf(p, 0.0f);
        }
#pragma unroll
        for (int f = 0; f < 5; ++f) dst[(size_t)f * NROWS + r] = vput[f];
      }
    }

    // stage this wave's 32 rows into LDS (lane == row)
#pragma unroll
    for (int f = 0; f < 5; ++f) zb[wv][lane][f] = vput[f];
    asm volatile("s_wait_dscnt 0x0" ::: "memory");  // wave-local LDS exchange

    // 8 WMMAs: S += Z^T Z over 4 rows each (A-frag == B-frag, symmetric GEMM)
#pragma unroll
    for (int g4 = 0; g4 < 8; ++g4) {
      const int r0 = g4 * 4 + 2 * hi;
      v2f frag;
      frag.x = zb[wv][r0 + 0][lm];
      frag.y = zb[wv][r0 + 1][lm];
      acc = __builtin_amdgcn_wmma_f32_16x16x4_f32(
          false, frag, false, frag, (short)0, acc, false, false);
    }
  }

  // D layout: lane l, vgpr g holds D[m=g+8*(l>=16)][n=l%16]; useful 6x6 block
  // lives in lanes 0..5, vgprs 0..5.
  if (lane < 6) {
#pragma unroll
    for (int g = 0; g < 6; ++g) wsum[wv][g * 6 + lane] = acc[g];
  }
  __syncthreads();
  if (tid < 36) {
    float s = 0.0f;
#pragma unroll
    for (int w = 0; w < 8; ++w) s += wsum[w][tid];   // fixed order
    partials[(size_t)blockIdx.x * 36 + tid] = s;
  }
}

// -----------------------------------------------------------------------------
// Finalize: reduce per-WG partials (deterministic, double), then fold
// BN(train) into the next linear:  y=(W a+b),  W'=W*g*rsqrt(var+eps),
// b'=(b-mu_y)*g*rsqrt(var+eps)+beta, with mu_y/var derived exactly from the
// moments of a.  outP layout: W' [out*5], then b' [out].
// -----------------------------------------------------------------------------
__global__ __launch_bounds__(512) void finalize_kernel(
    const float* __restrict__ partials,
    const float* __restrict__ W, const float* __restrict__ b,
    const float* __restrict__ g, const float* __restrict__ beta,
    int out_dim, float* __restrict__ outP)
{
  __shared__ double sred[8][36];
  __shared__ double S[36];
  const int tid = threadIdx.x;
  const int grp = tid / 36, comp = tid % 36;
  if (grp < 8) {
    double s = 0.0;
    for (int p = grp; p < NWG; p += 8) s += (double)partials[(size_t)p * 36 + comp];
    sred[grp][comp] = s;
  }
  __syncthreads();
  if (tid < 36) {
    double s = 0.0;
    for (int w = 0; w < 8; ++w) s += sred[w][tid];   // fixed order
    S[tid] = s;
  }
  __syncthreads();
  if (tid == 0) {
    const double invN = 1.0 / (double)NROWS;
    double mu[5], Exx[5][5];
    for (int j = 0; j < 5; ++j) mu[j] = S[j * 6 + 5] * invN;
    for (int j = 0; j < 5; ++j)
      for (int k = 0; k < 5; ++k) Exx[j][k] = S[j * 6 + k] * invN;
    for (int i = 0; i < out_dim; ++i) {
      double wm = 0.0;
      for (int j = 0; j < 5; ++j) wm += (double)W[i * 5 + j] * mu[j];
      const double bi  = (double)b[i];
      const double muy = wm + bi;
      double Eyy = bi * bi + 2.0 * bi * wm;
      for (int j = 0; j < 5; ++j)
        for (int k = 0; k < 5; ++k)
          Eyy += (double)W[i * 5 + j] * (double)W[i * 5 + k] * Exx[j][k];
      const double var = Eyy - muy * muy;
      const double sc  = (double)g[i] / sqrt(var + 1e-5);
      for (int j = 0; j < 5; ++j)
        outP[i * 5 + j] = (float)((double)W[i * 5 + j] * sc);
      outP[out_dim * 5 + i] = (float)((bi - muy) * sc + (double)beta[i]);
    }
  }
}

// -----------------------------------------------------------------------------
// Final head: z9 = relu(A9*z4 + c9) (2 feats, BN folded), out = lin10(z9).
// -----------------------------------------------------------------------------
__global__ __launch_bounds__(TPB) void final_kernel(
    const float* __restrict__ src, const float* __restrict__ p9,
    const float* __restrict__ w10, const float* __restrict__ b10,
    float* __restrict__ out)
{
  const size_t r = (size_t)blockIdx.x * TPB + threadIdx.x;
  float a[5];
#pragma unroll
  for (int f = 0; f < 5; ++f) a[f] = src[(size_t)f * NROWS + r];
  float t0 = p9[10], t1 = p9[11];
#pragma unroll
  for (int j = 0; j < 5; ++j) {
    t0 = fmaf(p9[j], a[j], t0);
    t1 = fmaf(p9[5 + j], a[j], t1);
  }
  t0 = fmaxf(t0, 0.0f);
  t1 = fmaxf(t1, 0.0f);
  out[r] = fmaf(w10[0], t0, fmaf(w10[1], t1, b10[0]));
}

extern "C" void kernel_launch(void* const* d_in, const int* in_sizes, int n_in,
                              void* d_out, int out_size, void* d_ws, size_t ws_size,
                              hipStream_t stream)
{
  (void)in_sizes; (void)n_in; (void)out_size; (void)ws_size;
  const float* x    = (const float*)d_in[0];
  const float* lw   = (const float*)d_in[1];
  const float* lb   = (const float*)d_in[2];
  const float* sw   = (const float*)d_in[3];
  const float* sb   = (const float*)d_in[4];
  const float* bg   = (const float*)d_in[5];
  const float* bb   = (const float*)d_in[6];
  const float* l9w  = (const float*)d_in[7];
  const float* l9b  = (const float*)d_in[8];
  const float* b9g  = (const float*)d_in[9];
  const float* b9b  = (const float*)d_in[10];
  const float* l10w = (const float*)d_in[11];
  const float* l10b = (const float*)d_in[12];

  float* wsf      = (float*)d_ws;
  float* partials = wsf;                       // 36 * NWG floats
  float* pslots   = wsf + 36 * NWG;            // 9 slots x 32 floats
  float* zbuf     = pslots + 9 * 32;           // 5 * NROWS floats (SoA, in-place)
  float* PH[4], *PZ[4];
  for (int k = 0; k < 4; ++k) { PH[k] = pslots + k * 32; PZ[k] = pslots + (4 + k) * 32; }
  float* P9 = pslots + 8 * 32;

  const dim3 grd(NWG), blk(TPB);

  // moments of x -> folded affine for lin0/bn0
  stage_kernel<0, true><<<grd, blk, 0, stream>>>(x, nullptr, nullptr, nullptr,
                                                 nullptr, nullptr, partials);
  finalize_kernel<<<1, 512, 0, stream>>>(partials, lw, lb, bg, bb, 5, PH[0]);

  const float* cur = x;
  for (int k = 0; k < 4; ++k) {
    // h-pass: moments of h_k
    if (k == 0)
      stage_kernel<1, true><<<grd, blk, 0, stream>>>(cur, nullptr, PH[k], nullptr,
                                                     nullptr, nullptr, partials);
    else
      stage_kernel<1, false><<<grd, blk, 0, stream>>>(cur, nullptr, PH[k], nullptr,
                                                      nullptr, nullptr, partials);
    finalize_kernel<<<1, 512, 0, stream>>>(partials, lw + (2 * k + 1) * 25,
                                           lb + (2 * k + 1) * 5, bg + (2 * k + 1) * 5,
                                           bb + (2 * k + 1) * 5, 5, PZ[k]);
    // z-pass: write z_k (SoA; in-place over zbuf for k>=1), moments of z_k
    if (k == 0)
      stage_kernel<2, true><<<grd, blk, 0, stream>>>(cur, zbuf, PH[k], PZ[k],
                                                     sw + 25 * k, sb + 5 * k, partials);
    else
      stage_kernel<2, false><<<grd, blk, 0, stream>>>(cur, zbuf, PH[k], PZ[k],
                                                      sw + 25 * k, sb + 5 * k, partials);
    if (k < 3)
      finalize_kernel<<<1, 512, 0, stream>>>(partials, lw + (2 * k + 2) * 25,
                                             lb + (2 * k + 2) * 5, bg + (2 * k + 2) * 5,
                                             bb + (2 * k + 2) * 5, 5, PH[k + 1]);
    else
      finalize_kernel<<<1, 512, 0, stream>>>(partials, l9w, l9b, b9g, b9b, 2, P9);
    cur = zbuf;
  }

  final_kernel<<<dim3(NROWS / TPB), blk, 0, stream>>>(cur, P9, l10w, l10b,
                                                      (float*)d_out);
}